// get_patches_64037962383753
// MI455X (gfx1250) — compile-verified
//
#include <hip/hip_runtime.h>
#include <math.h>

// ---- problem constants (from reference) ----
#define BATCH 8
#define NPTS  16384
#define MCEN  1024
#define KNN   32
#define COLS_PER_ROUND 256                 // 8 waves * 2 tiles * 16 cols
#define ROUNDS (NPTS / COLS_PER_ROUND)     // 64

typedef float v2f __attribute__((ext_vector_type(2)));
typedef float v4f __attribute__((ext_vector_type(4)));
typedef float v8f __attribute__((ext_vector_type(8)));

// Workgroup: 256 threads = 8 waves (wave32). Each workgroup owns one batch b
// and 16 consecutive centroids [m0, m0+16). Each wave computes two 16x16 score
// tiles per round with V_WMMA_F32_16X16X4_F32; the block cooperatively
// maintains a per-row top-32 (smallest score) set in LDS with a rejection
// threshold that makes steady-state cost ~1 compare per score.
__global__ __launch_bounds__(256)
void knn_patches_kernel(const float* __restrict__ x0,   // [B][N][3]
                        const float* __restrict__ x1,   // [B][M][3]
                        float* __restrict__ outFeat,    // [B][7][M][K]
                        float* __restrict__ outCen,     // [B][3][M]
                        int*   __restrict__ outIdx)     // [B][M][K]
{
    const int b   = blockIdx.x >> 6;          // 64 m-tiles per batch
    const int m0  = (blockIdx.x & 63) << 4;   // 16 centroids per tile
    const int tid = threadIdx.x;
    const int waveId = tid >> 5;              // 0..7
    const int lane   = tid & 31;
    const int half   = lane >> 4;             // 0: lanes 0-15, 1: lanes 16-31
    const int l15    = lane & 15;

    // ---- LDS state ----
    __shared__ float topS[16 * KNN];             // current top-32 scores per row
    __shared__ int   topI[16 * KNN];             // matching column indices
    __shared__ float candS[16 * COLS_PER_ROUND]; // per-round candidate lists
    __shared__ int   candI[16 * COLS_PER_ROUND];
    __shared__ int   cnt[16];                    // candidates appended this round
    __shared__ int   tcnt[16];                   // valid entries in top buffer
    __shared__ __align__(16) float thresh[16];   // 32nd-best score (+inf until full)

    if (tid < 16) {
        cnt[tid]    = 0;
        tcnt[tid]   = 0;
        thresh[tid] = INFINITY;
    }
    __syncthreads();

    // ---- Build A matrix (16x4 f32): row m = centroid, K = (x, y, z, 1) ----
    // A layout: lanes 0-15 hold {K=0,K=1}, lanes 16-31 hold {K=2,K=3}.
    const float* cenP = x1 + ((size_t)b * MCEN + (m0 + l15)) * 3;
    const float c0 = cenP[0], c1 = cenP[1], c2 = cenP[2];
    v2f amat;
    amat.x = half ? c2   : c0;
    amat.y = half ? 1.0f : c1;

    const float* x0b = x0 + (size_t)b * NPTS * 3;

    // software-pipelined point loads: this wave's two columns per round
    const int colBase = waveId * 32 + l15;       // + it*256; second tile at +16
    int col0 = colBase;
    int col1 = colBase + 16;
    float pA0 = x0b[col0 * 3 + 0], pA1 = x0b[col0 * 3 + 1], pA2 = x0b[col0 * 3 + 2];
    float pB0 = x0b[col1 * 3 + 0], pB1 = x0b[col1 * 3 + 1], pB2 = x0b[col1 * 3 + 2];

    for (int it = 0; it < ROUNDS; ++it) {
        // ---- prefetch next round's points (clamped redundant load on last) ----
        const int itn = (it + 1 < ROUNDS) ? (it + 1) : it;
        const int ncol0 = itn * COLS_PER_ROUND + colBase;
        const int ncol1 = ncol0 + 16;
        const float qA0 = x0b[ncol0 * 3 + 0], qA1 = x0b[ncol0 * 3 + 1], qA2 = x0b[ncol0 * 3 + 2];
        const float qB0 = x0b[ncol1 * 3 + 0], qB1 = x0b[ncol1 * 3 + 1], qB2 = x0b[ncol1 * 3 + 2];

        // ---- per-lane row thresholds: two b128 LDS loads ----
        const v4f* tp = (const v4f*)(thresh + (half << 3));
        const v4f t0 = tp[0];
        const v4f t1 = tp[1];

        // ---- B matrices (4x16 f32): col n = point, K = (-2px,-2py,-2pz,|p|^2) ----
        v2f bmatA, bmatB;
        bmatA.x = half ? (-2.0f * pA2) : (-2.0f * pA0);
        bmatA.y = half ? (pA0 * pA0 + pA1 * pA1 + pA2 * pA2) : (-2.0f * pA1);
        bmatB.x = half ? (-2.0f * pB2) : (-2.0f * pB0);
        bmatB.y = half ? (pB0 * pB0 + pB1 * pB1 + pB2 * pB2) : (-2.0f * pB1);

        v8f acc0 = {}, acc1 = {};
        acc0 = __builtin_amdgcn_wmma_f32_16x16x4_f32(
            false, amat, false, bmatA, (short)0, acc0, false, false);
        acc1 = __builtin_amdgcn_wmma_f32_16x16x4_f32(
            false, amat, false, bmatB, (short)0, acc1, false, false);

        // D layout: lane<16 holds D[v][lane], lane>=16 holds D[v+8][lane-16].
        #pragma unroll
        for (int v = 0; v < 8; ++v) {
            const float th = (v < 4) ? t0[v] : t1[v - 4];
            const int   r  = v + (half << 3);
            const float s0 = acc0[v];
            if (s0 < th) {
                const int pos = atomicAdd(&cnt[r], 1);    // bounded by 256/row/round
                candS[r * COLS_PER_ROUND + pos] = s0;
                candI[r * COLS_PER_ROUND + pos] = col0;
            }
            const float s1 = acc1[v];
            if (s1 < th) {
                const int pos = atomicAdd(&cnt[r], 1);
                candS[r * COLS_PER_ROUND + pos] = s1;
                candI[r * COLS_PER_ROUND + pos] = col1;
            }
        }
        __syncthreads();

        // ---- Merge: wave w refreshes rows 2w and 2w+1 ----
        for (int rr = 0; rr < 2; ++rr) {
            const int r = waveId * 2 + rr;
            const int c = cnt[r];
            if (c > 0) {
                const int t = tcnt[r];
                const int V = t + c;                      // <= 32 + 256 = 288
                float myS[9]; int myI[9]; int myR[9]; int own = 0;
                for (int i = lane; i < V; i += 32) {
                    const float si = (i < t) ? topS[r * KNN + i]
                                             : candS[r * COLS_PER_ROUND + (i - t)];
                    const int   ii = (i < t) ? topI[r * KNN + i]
                                             : candI[r * COLS_PER_ROUND + (i - t)];
                    int rank = 0;
                    for (int j = 0; j < V; ++j) {
                        const float sj = (j < t) ? topS[r * KNN + j]
                                                 : candS[r * COLS_PER_ROUND + (j - t)];
                        const int   ij = (j < t) ? topI[r * KNN + j]
                                                 : candI[r * COLS_PER_ROUND + (j - t)];
                        rank += (sj < si) || (sj == si && ij < ii);
                    }
                    myS[own] = si; myI[own] = ii; myR[own] = rank; ++own;
                }
                // write phase (all LDS reads above retire first: lockstep wave)
                for (int q = 0; q < own; ++q) {
                    if (myR[q] < KNN) {
                        topS[r * KNN + myR[q]] = myS[q];
                        topI[r * KNN + myR[q]] = myI[q];
                    }
                    if (myR[q] == KNN - 1) thresh[r] = myS[q];  // new 32nd best
                }
                if (lane == 0) {
                    tcnt[r] = (V < KNN) ? V : KNN;
                    cnt[r]  = 0;
                }
            }
        }
        __syncthreads();

        // rotate pipeline registers
        col0 = ncol0; col1 = ncol1;
        pA0 = qA0; pA1 = qA1; pA2 = qA2;
        pB0 = qB0; pB1 = qB1; pB2 = qB2;
    }

    // ---- Output phase ----
    // Each wave handles rows 2w, 2w+1; lane owns top-buffer slot `lane`.
    const size_t chStride = (size_t)MCEN * KNN;   // 32768
    for (int rr = 0; rr < 2; ++rr) {
        const int r = waveId * 2 + rr;
        const int m = m0 + r;
        const float si = topS[r * KNN + lane];
        const int   ii = topI[r * KNN + lane];
        int rank = 0;
        for (int j = 0; j < KNN; ++j) {
            const float sj = topS[r * KNN + j];
            const int   ij = topI[r * KNN + j];
            rank += (sj < si) || (sj == si && ij < ii);
        }
        const float* pp = x0b + (size_t)ii * 3;
        const float* cp = x1 + ((size_t)b * MCEN + m) * 3;
        const float rx = pp[0] - cp[0];
        const float ry = pp[1] - cp[1];
        const float rz = pp[2] - cp[2];
        const float dist = sqrtf(rx * rx + ry * ry + rz * rz);

        const size_t base = (size_t)b * 7 * chStride + (size_t)m * KNN + rank;
        outFeat[base + 0 * chStride] = rx;
        outFeat[base + 1 * chStride] = ry;
        outFeat[base + 2 * chStride] = rz;
        outFeat[base + 3 * chStride] = dist;
        outFeat[base + 4 * chStride] = cp[0];
        outFeat[base + 5 * chStride] = cp[1];
        outFeat[base + 6 * chStride] = cp[2];
        outIdx[((size_t)b * MCEN + m) * KNN + rank] = ii;
    }

    // centroids output [B][3][M]
    if (tid < 48) {
        const int r = tid & 15;
        const int d = tid >> 4;
        outCen[((size_t)b * 3 + d) * MCEN + (m0 + r)] =
            x1[((size_t)b * MCEN + (m0 + r)) * 3 + d];
    }
}

extern "C" void kernel_launch(void* const* d_in, const int* in_sizes, int n_in,
                              void* d_out, int out_size, void* d_ws, size_t ws_size,
                              hipStream_t stream) {
    (void)in_sizes; (void)n_in; (void)d_ws; (void)ws_size; (void)out_size;
    const float* x0 = (const float*)d_in[0];   // [8,16384,3]
    const float* x1 = (const float*)d_in[1];   // [8,1024,3]

    float* outFeat = (float*)d_out;                              // 8*7*1024*32 = 1,835,008
    float* outCen  = outFeat + (size_t)BATCH * 7 * MCEN * KNN;   // 8*3*1024   =    24,576
    int*   outIdx  = (int*)(outCen + (size_t)BATCH * 3 * MCEN);  // 8*1024*32  =   262,144

    dim3 grid(BATCH * (MCEN / 16));   // 512 workgroups
    dim3 block(256);                  // 8 waves (wave32)
    knn_patches_kernel<<<grid, block, 0, stream>>>(x0, x1, outFeat, outCen, outIdx);
}